// SO2Net_43293270344160
// MI455X (gfx1250) — compile-verified
//
#include <hip/hip_runtime.h>
#include <cstdint>
#include <cstddef>

// ---------------------------------------------------------------------------
// SO(2)-equivariant net for MI455X (gfx1250, wave32).
// Fused complex GEMM on V_WMMA_F32_16X16X4_F32 with the 3-multiply Gauss
// trick: K1=(Ar+Ai)Wr, K2=Ar(Wi-Wr), K3=Ai(Wr+Wi); Cr=K1-K3, Ci=K1+K2.
// A-panel built branchlessly into de-interleaved LDS planes; weights are
// pre-transformed into 3 streaming planes. Working set is L2-resident.
// ---------------------------------------------------------------------------

constexpr int BATCH = 4096;
constexpr int MAXE  = 16;   // max entries per ky (dc[0]=14 for D=10)

// Packed entry: bits[11:0] = float offset of factor a within channel block,
// bits[23:12] = float offset of factor b, bit[24] = copy flag.
struct KyTab { int pk[MAXE]; int cnt; };

struct LayerTab {
  KyTab t[21];  // ky = -10..10 -> index ky+10
  constexpr void push(int ky, int offA, int offB, int copy) {
    KyTab& k = t[ky + 10];
    k.pk[k.cnt] = offA | (offB << 12) | (copy << 24);
    k.cnt = k.cnt + 1;
  }
  constexpr LayerTab(int D, int M) : t{} {
    const int Dp1 = D + 1;
    // channel block layout: [xp(0..D) cplx][xn(0..D) cplx], floats.
    // Mirrors the reference's append order into inter[ky].
    for (int i = 0; i <= D; ++i)
      for (int j = 0; j <= D; ++j) {
        push(-i + j, 2 * (Dp1 + i), 2 * j, 0);                    // xn_i*xp_j*0.5
        if (i <= j && i + j <= M) {
          push(i + j, 2 * i, 2 * j, 0);                           // xp_i*xp_j*0.5
          push(-i - j, 2 * (Dp1 + i), 2 * (Dp1 + j), 0);          // xn_i*xn_j*0.5
        }
      }
    for (int d = 0; d <= D; ++d) {
      push(d, 2 * d, 0, 1);                                       // copy xp_d
      if (d > 0) push(-d, 2 * (Dp1 + d), 0, 1);                   // copy xn_d
    }
  }
};

__device__ constexpr LayerTab TAB8(8, 10);
__device__ constexpr LayerTab TAB10(10, 10);
constexpr LayerTab HTAB8(8, 10);
constexpr LayerTab HTAB10(10, 10);
constexpr int E0_8  = HTAB8.t[10].cnt;   // 12
constexpr int E0_10 = HTAB10.t[10].cnt;  // 14

typedef __attribute__((ext_vector_type(2))) float v2f;
typedef __attribute__((ext_vector_type(8))) float v8f;

// Branchless complex feature from a packed table entry.
__device__ __forceinline__ void featP(const float* __restrict__ chanBase, int pk,
                                      float& fr, float& fi) {
  const int offA = pk & 0xFFF;
  const int offB = (pk >> 12) & 0xFFF;
  const bool cp  = (pk >> 24) & 1;
  const float2 fa = *(const float2*)(chanBase + offA);
  float2 fb = *(const float2*)(chanBase + offB);
  const float fbx = cp ? 2.0f : fb.x;     // 0.5*(a * (2+0i)) == a  (copy case)
  const float fby = cp ? 0.0f : fb.y;
  fr = 0.5f * (fa.x * fbx - fa.y * fby);
  fi = 0.5f * (fa.x * fby + fa.y * fbx);
}

struct WPtrs { const float* w[21]; };  // original weight pointer per projection p

// Pre-transform complex weights (Cout,Ktot) into 3 float planes per projection:
// plane0 = Wr, plane1 = Wi - Wr, plane2 = Wr + Wi. wt stride per p = strideKy.
__global__ void wtrans_kernel(WPtrs wp, float* __restrict__ wt,
                              int Cout, int Cin, int od, int tabsel, int strideKy) {
  const int p  = blockIdx.y;
  const int ky = (p <= od) ? p : -(p - od);
  const KyTab& kt = tabsel ? TAB10.t[ky + 10] : TAB8.t[ky + 10];
  const int Ktot = Cin * kt.cnt;
  const int tot  = Cout * Ktot;
  const int idx  = blockIdx.x * blockDim.x + threadIdx.x;
  if (idx >= tot) return;
  const float* __restrict__ w = wp.w[p];
  const float wr = w[2 * idx];
  const float wi = w[2 * idx + 1];
  float* base = wt + (size_t)p * strideKy;
  base[idx]            = wr;
  base[tot + idx]      = wi - wr;
  base[2 * tot + idx]  = wr + wi;
}

// Block: 128 threads = 4 waves. One block -> 16(M batch) x 64(N Cout) complex
// tile for projection p; grid = (B/16, 1, 2*od+1). A staged in 2 LDS planes.
__global__ __launch_bounds__(128) void so2_gemm_kernel(
    const float* __restrict__ xin,   // (B, Cin, 2, Dp1) cplx
    const float* __restrict__ mlp0,  // (B, nin0) real — replaces feats for ky==0
    const float* __restrict__ wt,    // transformed weight planes
    int strideKy,
    float* __restrict__ out,         // (B, Cout, 2, od+1) cplx
    int Cin, int Cout, int Dp1, int od, int tabsel)
{
  extern __shared__ float lds[];     // plane0: Ar[16][rs], plane1: Ai[16][rs]

  const int p  = blockIdx.z;
  const int ky = (p <= od) ? p : -(p - od);
  const KyTab& kt = tabsel ? TAB10.t[ky + 10] : TAB8.t[ky + 10];
  const int E    = kt.cnt;           // 4..14
  const int Ktot = Cin * E;
  const int Kpad = (Ktot + 3) & ~3;
  const int rs   = Kpad + 4;         // row stride (floats); %64==4 bank skew
  const bool useMlp = (ky == 0);
  const int tid   = threadIdx.x;
  const int mbase = blockIdx.x * 16;

  // ---- Cooperative A-panel fill (de-interleaved planes, zero-padded) ----
  {
    const int row  = tid >> 3;           // 16 rows x 8 k-lanes
    const int mrow = mbase + row;
    int k = tid & 7;
    int c = 0, e = k;
    while (e >= E) { e -= E; ++c; }
    const size_t chanStride = (size_t)4 * Dp1;   // floats per (row,channel) block
    const float* mrowp = mlp0 + (size_t)mrow * Ktot;
    float* aR = lds + (size_t)row * rs;
    float* aI = lds + (size_t)(16 + row) * rs;
    for (; k < Kpad; k += 8) {
      float fr = 0.f, fi = 0.f;
      if (k < Ktot) {
        if (useMlp) {
          fr = mrowp[k];
        } else {
          const float* cb = xin + chanStride * ((size_t)mrow * Cin + c);
          featP(cb, kt.pk[e], fr, fi);
        }
      }
      aR[k] = fr;                        // ds_store_b32 x2
      aI[k] = fi;
      e += 8; while (e >= E) { e -= E; ++c; }
    }
  }
  __syncthreads();

  // ---- WMMA compute: each wave owns one 16-wide N tile ----
  const int lane = tid & 31;
  const int wv   = tid >> 5;
  const int lh   = lane >> 4;            // K-pair selector per WMMA layout
  const int m15  = lane & 15;
  const int ncol = wv * 16 + m15;
  const int nc   = (ncol < Cout) ? ncol : (Cout - 1);    // clamp for addressing
  const int tot  = Cout * Ktot;
  const float* wbase = wt + (size_t)p * strideKy + (size_t)nc * Ktot;
  const float* pW1 = wbase + 2 * lh;             // Wr
  const float* pW2 = pW1 + tot;                  // Wi - Wr
  const float* pW3 = pW2 + tot;                  // Wr + Wi
  const float* aR  = lds + (size_t)m15 * rs + 2 * lh;
  const float* aI  = aR + (size_t)16 * rs;
  __builtin_prefetch(pW1, 0, 1);                 // global_prefetch_b8

  v8f k1 = {0.f,0.f,0.f,0.f,0.f,0.f,0.f,0.f};
  v8f k2 = {0.f,0.f,0.f,0.f,0.f,0.f,0.f,0.f};
  v8f k3 = {0.f,0.f,0.f,0.f,0.f,0.f,0.f,0.f};

  // Main loop: valid when Ktot even (always for Cin=64; most layer-0 kys).
  const int Kmain = (Ktot & 1) ? 0 : (Ktot & ~3);
#pragma unroll 4
  for (int k0 = 0; k0 < Kmain; k0 += 4) {
    const v2f ar = *(const v2f*)(aR + k0);       // ds_load_b64
    const v2f ai = *(const v2f*)(aI + k0);       // ds_load_b64
    const v2f as = ar + ai;
    const v2f w1 = *(const v2f*)(pW1 + k0);      // global_load_b64
    const v2f w2 = *(const v2f*)(pW2 + k0);
    const v2f w3 = *(const v2f*)(pW3 + k0);
    k1 = __builtin_amdgcn_wmma_f32_16x16x4_f32(false, as, false, w1, (short)0, k1, false, false);
    k2 = __builtin_amdgcn_wmma_f32_16x16x4_f32(false, ar, false, w2, (short)0, k2, false, false);
    k3 = __builtin_amdgcn_wmma_f32_16x16x4_f32(false, ai, false, w3, (short)0, k3, false, false);
  }
  // Tail (odd-K layer-0 projections): A from zero-padded LDS, B guarded scalar.
  for (int k0 = Kmain; k0 < Ktot; k0 += 4) {
    const int ka = k0 + 2 * lh;
    const int kb = ka + 1;
    const v2f ar = *(const v2f*)(aR + k0);
    const v2f ai = *(const v2f*)(aI + k0);
    const v2f as = ar + ai;
    float w1x = 0.f, w1y = 0.f, w2x = 0.f, w2y = 0.f, w3x = 0.f, w3y = 0.f;
    if (ka < Ktot) { w1x = wbase[ka]; w2x = wbase[tot + ka]; w3x = wbase[2*tot + ka]; }
    if (kb < Ktot) { w1y = wbase[kb]; w2y = wbase[tot + kb]; w3y = wbase[2*tot + kb]; }
    v2f w1; w1.x = w1x; w1.y = w1y;
    v2f w2; w2.x = w2x; w2.y = w2y;
    v2f w3; w3.x = w3x; w3.y = w3y;
    k1 = __builtin_amdgcn_wmma_f32_16x16x4_f32(false, as, false, w1, (short)0, k1, false, false);
    k2 = __builtin_amdgcn_wmma_f32_16x16x4_f32(false, ar, false, w2, (short)0, k2, false, false);
    k3 = __builtin_amdgcn_wmma_f32_16x16x4_f32(false, ai, false, w3, (short)0, k3, false, false);
  }

  const v8f accr = k1 - k3;   // Cr
  const v8f acci = k1 + k2;   // Ci

  // C/D layout: VGPR v, lanes0-15 -> M=v, lanes16-31 -> M=v+8; N=lane&15.
  const int s  = (p <= od) ? 0 : 1;
  const int ad = (p <= od) ? p : (p - od);
  if (ncol < Cout) {
    for (int v = 0; v < 8; ++v) {
      const int row = mbase + v + 8 * lh;
      size_t off = 2 * ((((size_t)row * Cout + ncol) * 2 + s) * (size_t)(od + 1) + ad);
      out[off]     = accr[v];
      out[off + 1] = acci[v];
    }
  }
}

// xf = x @ F^T ; output (B, 1, 2, 9) cplx with s=0: xf, s=1: conj(xf).
__global__ void xf_kernel(const float* __restrict__ x, const float* __restrict__ F,
                          float* __restrict__ act0) {
  int b = blockIdx.x * blockDim.x + threadIdx.x;
  if (b >= BATCH) return;
  float xr[9];
  for (int m = 0; m < 9; ++m) xr[m] = x[b * 9 + m];
  float* o = act0 + (size_t)b * 36;
  for (int k = 0; k < 9; ++k) {
    float sr = 0.f, si = 0.f;
    for (int m = 0; m < 9; ++m) {
      sr += xr[m] * F[2 * (k * 9 + m)];
      si += xr[m] * F[2 * (k * 9 + m) + 1];
    }
    o[2*k] = sr; o[2*k+1] = si;             // xp
    o[18 + 2*k] = sr; o[18 + 2*k+1] = -si;  // xn = conj
  }
}

// Real part of the ky==0 intermediate features -> MLP input (B, nin0).
__global__ void feats0_kernel(const float* __restrict__ xin, float* __restrict__ r0,
                              int Cin, int Dp1, int tabsel) {
  const KyTab& kt = tabsel ? TAB10.t[10] : TAB8.t[10];
  const int E = kt.cnt;
  const int nin0 = Cin * E;
  int idx = blockIdx.x * blockDim.x + threadIdx.x;
  if (idx >= BATCH * nin0) return;
  int b = idx / nin0;
  int r = idx - b * nin0;
  int c = r / E;
  int e = r - c * E;
  const float* cb = xin + (size_t)4 * Dp1 * ((size_t)b * Cin + c);
  float fr, fi;
  featP(cb, kt.pk[e], fr, fi);
  (void)fi;
  r0[idx] = fr;
}

// nin0 -> 10 -> 10 -> nin0 MLP (sigmoid on hidden layers). One thread per row.
__global__ void mlp_kernel(const float* __restrict__ r0,
                           const float* __restrict__ W0, const float* __restrict__ B0,
                           const float* __restrict__ W1, const float* __restrict__ B1,
                           const float* __restrict__ W2, const float* __restrict__ B2,
                           float* __restrict__ m0, int nin0) {
  int b = blockIdx.x * blockDim.x + threadIdx.x;
  if (b >= BATCH) return;
  float h1[10];
  for (int o = 0; o < 10; ++o) h1[o] = B0[o];
  const float* rrow = r0 + (size_t)b * nin0;
  for (int k = 0; k < nin0; ++k) {
    float v = rrow[k];
    for (int o = 0; o < 10; ++o) h1[o] += W0[(size_t)o * nin0 + k] * v;
  }
  for (int o = 0; o < 10; ++o) h1[o] = 1.f / (1.f + __expf(-h1[o]));
  float h2[10];
  for (int o = 0; o < 10; ++o) {
    float s = B1[o];
    for (int k = 0; k < 10; ++k) s += W1[o * 10 + k] * h1[k];
    h2[o] = 1.f / (1.f + __expf(-s));
  }
  float* orow = m0 + (size_t)b * nin0;
  for (int k = 0; k < nin0; ++k) {
    float s = B2[k];
    for (int o = 0; o < 10; ++o) s += W2[(size_t)k * 10 + o] * h2[o];
    orow[k] = s;
  }
}

// Zero the (s=1, a=0) slice the GEMM never writes.
__global__ void zero_neg0_kernel(float* __restrict__ out, int Cout, int od) {
  int idx = blockIdx.x * blockDim.x + threadIdx.x;
  if (idx >= BATCH * Cout) return;
  size_t off = 2 * (((size_t)idx * 2 + 1) * (size_t)(od + 1));
  out[off] = 0.f; out[off + 1] = 0.f;
}

// ---------------------------------------------------------------------------
// d_in flatten order (jax sorted-dict): [F, layer0{mlp W0,b0,W1,b1,W2,b2;
// w_ky string-sorted}, layer1, layer2, x]  => F=0, layers at 1+27*l, x=82.
// ---------------------------------------------------------------------------
extern "C" void kernel_launch(void* const* d_in, const int* in_sizes, int n_in,
                              void* d_out, int out_size, void* d_ws, size_t ws_size,
                              hipStream_t stream) {
  (void)in_sizes; (void)n_in; (void)out_size; (void)ws_size;
  const float* F = (const float*)d_in[0];
  const float* x = (const float*)d_in[82];

  // string-sorted w_ky keys -> ky values, in d_in order
  static const int kyOrder[21] = {-1,-10,-2,-3,-4,-5,-6,-7,-8,-9,0,1,10,2,3,4,5,6,7,8,9};

  float* ws = (float*)d_ws;
  const size_t ACT      = (size_t)BATCH * 64 * 2 * 11 * 2;  // 11.53M floats
  const size_t MLPBUF   = (size_t)BATCH * 896;
  const int    strideKy = 3 * 64 * 896;                      // floats per p (max)
  float* act0 = ws;                                // (B,1,2,9) cplx: B*36 floats
  float* actA = ws + (size_t)BATCH * 36;
  float* actB = actA + ACT;
  float* r0   = actB + ACT;
  float* m0   = r0 + MLPBUF;
  float* wtb  = m0 + MLPBUF;                       // 21*strideKy = 3.61M floats

  // Allow large dynamic LDS (112.5KB A-panel for layers 1/2).
  hipFuncSetAttribute((const void*)so2_gemm_kernel,
                      hipFuncAttributeMaxDynamicSharedMemorySize, 120 * 1024);

  xf_kernel<<<dim3((BATCH + 255) / 256), dim3(256), 0, stream>>>(x, F, act0);

  struct LayerCfg { int Cin, Cout, Dp1, od, tabsel; const float* in; float* out; };
  const LayerCfg cfg[3] = {
    { 1, 64,  9, 10, 0, act0, actA },
    {64, 64, 11, 10, 1, actA, actB },
    {64,  9, 11, 10, 1, actB, (float*)d_out },
  };

  for (int l = 0; l < 3; ++l) {
    const int base = 1 + 27 * l;
    const float* W0 = (const float*)d_in[base + 0];
    const float* B0 = (const float*)d_in[base + 1];
    const float* W1 = (const float*)d_in[base + 2];
    const float* B1 = (const float*)d_in[base + 3];
    const float* W2 = (const float*)d_in[base + 4];
    const float* B2 = (const float*)d_in[base + 5];

    WPtrs wp;
    for (int pj = 0; pj < 21; ++pj) {
      const int ky = (pj <= 10) ? pj : -(pj - 10);
      const float* wptr = nullptr;
      for (int idx = 0; idx < 21; ++idx)
        if (kyOrder[idx] == ky) { wptr = (const float*)d_in[base + 6 + idx]; break; }
      wp.w[pj] = wptr;
    }

    const LayerCfg& L = cfg[l];
    const int Emax  = L.tabsel ? 14 : 12;
    const int E0    = L.tabsel ? E0_10 : E0_8;
    const int nin0  = L.Cin * E0;
    const int tot0  = BATCH * nin0;
    const int KpadM = (L.Cin * Emax + 3) & ~3;
    const size_t ldsBytes = (size_t)32 * (KpadM + 4) * sizeof(float);
    const int totW  = L.Cout * L.Cin * Emax;       // upper bound on Cout*Ktot

    wtrans_kernel<<<dim3((totW + 255) / 256, 2 * L.od + 1), dim3(256), 0, stream>>>(
        wp, wtb, L.Cout, L.Cin, L.od, L.tabsel, strideKy);
    feats0_kernel<<<dim3((tot0 + 255) / 256), dim3(256), 0, stream>>>(
        L.in, r0, L.Cin, L.Dp1, L.tabsel);
    mlp_kernel<<<dim3((BATCH + 127) / 128), dim3(128), 0, stream>>>(
        r0, W0, B0, W1, B1, W2, B2, m0, nin0);
    so2_gemm_kernel<<<dim3(BATCH / 16, 1, 2 * L.od + 1), dim3(128), ldsBytes, stream>>>(
        L.in, m0, wtb, strideKy, L.out, L.Cin, L.Cout, L.Dp1, L.od, L.tabsel);
    zero_neg0_kernel<<<dim3((BATCH * L.Cout + 255) / 256), dim3(256), 0, stream>>>(
        L.out, L.Cout, L.od);
  }
}